// KGAT_64330020159802
// MI455X (gfx1250) — compile-verified
//
#include <hip/hip_runtime.h>
#include <math.h>

typedef __attribute__((ext_vector_type(2))) float v2f;
typedef __attribute__((ext_vector_type(8))) float v8f;

constexpr int N_USERS    = 50000;
constexpr int N_ENTITIES = 100000;
constexpr int N_NODES    = N_USERS + N_ENTITIES;   // 150000 = 9375 * 16
constexpr int N_EDGES    = 3000000;
constexpr int EMB_DIM    = 64;
constexpr int BATCH      = 8192;
constexpr float LEAKY    = 0.2f;

// ---------------------------------------------------------------- zero fill
__global__ __launch_bounds__(256) void zero4_kernel(float4* __restrict__ p, long n4) {
    long i = (long)blockIdx.x * blockDim.x + threadIdx.x;
    if (i < n4) p[i] = make_float4(0.f, 0.f, 0.f, 0.f);
}

// ------------------------------------------------- edge gather * A -> scatter
// side[h] += A[e] * ego[t]   (L2-resident atomics; float4 gather per thread)
__device__ inline void atomic_add_f32(float* p, float v) {
    __hip_atomic_fetch_add(p, v, __ATOMIC_RELAXED, __HIP_MEMORY_SCOPE_AGENT);
}

template<int DIN, bool CONCAT>
__global__ __launch_bounds__(256) void edge_scatter(
    const int* __restrict__ all_h, const int* __restrict__ all_t,
    const float* __restrict__ av,
    const float* __restrict__ ego,
    const float* __restrict__ uemb, const float* __restrict__ eemb,
    float* __restrict__ side)
{
    constexpr int CH = DIN / 4;                       // float4 chunks per edge
    constexpr int SH = (CH == 16) ? 4 : (CH == 8 ? 3 : 2);
    long idx = (long)blockIdx.x * blockDim.x + threadIdx.x;
    if (idx >= (long)N_EDGES * CH) return;
    int e = (int)(idx >> SH);
    int c = (int)(idx & (CH - 1));
    int t = all_t[e];
    int h = all_h[e];
    float a = av[e];
    const float* erow;
    if constexpr (CONCAT) {
        erow = (t < N_USERS) ? (uemb + (size_t)t * EMB_DIM)
                             : (eemb + (size_t)(t - N_USERS) * EMB_DIM);
    } else {
        erow = ego + (size_t)t * DIN;
    }
    const float4 v = *(const float4*)(erow + c * 4);
    float* dst = side + (size_t)h * DIN + c * 4;
    atomic_add_f32(dst + 0, a * v.x);
    atomic_add_f32(dst + 1, a * v.y);
    atomic_add_f32(dst + 2, a * v.z);
    atomic_add_f32(dst + 3, a * v.w);
}

// ---------------------------------------------------------- WMMA node phase
// per wave: 16 nodes x 16 out-cols tile.
//   sum_emb = leaky((ego+side) @ Wgc + bgc)
//   bi_emb  = leaky((ego*side) @ Wbi + bbi)
//   out     = sum_emb + bi_emb            (unnormalized; norm fused at gather)
template<int DIN, int DOUT, bool CONCAT>
__global__ __launch_bounds__(256) void gemm_bi(
    const float* __restrict__ ego,
    const float* __restrict__ uemb, const float* __restrict__ eemb,
    const float* __restrict__ side,
    const float* __restrict__ Wgc, const float* __restrict__ Bgc,
    const float* __restrict__ Wbi, const float* __restrict__ Bbi,
    float* __restrict__ out)
{
    constexpr int MTILES = N_NODES / 16;              // 9375
    int wave = (int)((blockIdx.x * blockDim.x + threadIdx.x) >> 5);
    if (wave >= MTILES) return;                       // wave-uniform exit
    int lane = threadIdx.x & 31;
    int half = lane >> 4;                             // k-pair selector
    int m    = lane & 15;                             // row (A) / col (B,D)
    int node = wave * 16 + m;

    const float* erow;
    if constexpr (CONCAT) {
        erow = (node < N_USERS) ? (uemb + (size_t)node * EMB_DIM)
                                : (eemb + (size_t)(node - N_USERS) * EMB_DIM);
    } else {
        erow = ego + (size_t)node * DIN;
    }
    const float* srow = side + (size_t)node * DIN;

    // hoist A fragments for both matmuls across full K
    v2f agc[DIN / 4], abi[DIN / 4];
#pragma unroll
    for (int kb = 0; kb < DIN / 4; ++kb) {
        int k = kb * 4 + half * 2;
        float e0 = erow[k], e1 = erow[k + 1];
        float s0 = srow[k], s1 = srow[k + 1];
        agc[kb][0] = e0 + s0; agc[kb][1] = e1 + s1;
        abi[kb][0] = e0 * s0; abi[kb][1] = e1 * s1;
    }

#pragma unroll
    for (int nt = 0; nt < DOUT / 16; ++nt) {
        int col = nt * 16 + m;
        v8f dgc = {};
        v8f dbi = {};
#pragma unroll
        for (int kb = 0; kb < DIN / 4; ++kb) {
            int k = kb * 4 + half * 2;
            v2f bg, bb;
            bg[0] = Wgc[(size_t)k       * DOUT + col];
            bg[1] = Wgc[(size_t)(k + 1) * DOUT + col];
            bb[0] = Wbi[(size_t)k       * DOUT + col];
            bb[1] = Wbi[(size_t)(k + 1) * DOUT + col];
            dgc = __builtin_amdgcn_wmma_f32_16x16x4_f32(
                false, agc[kb], false, bg, (short)0, dgc, false, false);
            dbi = __builtin_amdgcn_wmma_f32_16x16x4_f32(
                false, abi[kb], false, bb, (short)0, dbi, false, false);
        }
        float bgs = Bgc[col];
        float bbs = Bbi[col];
#pragma unroll
        for (int i = 0; i < 8; ++i) {
            int row = i + half * 8;
            float x = dgc[i] + bgs;
            x = (x > 0.f) ? x : LEAKY * x;
            float y = dbi[i] + bbs;
            y = (y > 0.f) ? y : LEAKY * y;
            out[(size_t)(wave * 16 + row) * DOUT + col] = x + y;
        }
    }
}

// -------------------------------------------- final gather + fused row norms
// one 64-thread block per output row; concat [ego0 | n(ego1) | n(ego2) | n(ego3)]
__global__ __launch_bounds__(64) void gather_out_kernel(
    const int* __restrict__ users, const int* __restrict__ pos, const int* __restrict__ neg,
    const float* __restrict__ uemb, const float* __restrict__ eemb,
    const float* __restrict__ ego1, const float* __restrict__ ego2, const float* __restrict__ ego3,
    float* __restrict__ out)
{
    __shared__ float red[64];
    int slot  = blockIdx.x;                 // [0, 3*BATCH)
    int which = slot / BATCH;               // 0=users 1=pos 2=neg
    int i     = slot - which * BATCH;
    int idx   = (which == 0) ? users[i] : ((which == 1) ? pos[i] : neg[i]);
    int node  = (which == 0) ? idx : (N_USERS + idx);
    int t     = threadIdx.x;
    float* orow = out + (size_t)slot * 176;

    // segment 0: raw initial embeddings (64)
    const float* e0 = (node < N_USERS) ? (uemb + (size_t)node * 64)
                                       : (eemb + (size_t)(node - N_USERS) * 64);
    orow[t] = e0[t];

    // segment 1: ego1 / ||ego1||  (64)
    {
        float v = ego1[(size_t)node * 64 + t];
        red[t] = v * v; __syncthreads();
        for (int s = 32; s > 0; s >>= 1) { if (t < s) red[t] += red[t + s]; __syncthreads(); }
        float nrm = fmaxf(sqrtf(red[0]), 1e-12f);
        __syncthreads();
        orow[64 + t] = v / nrm;
    }
    // segment 2: ego2 / ||ego2||  (32)
    {
        float v = (t < 32) ? ego2[(size_t)node * 32 + t] : 0.f;
        red[t] = v * v; __syncthreads();
        for (int s = 32; s > 0; s >>= 1) { if (t < s) red[t] += red[t + s]; __syncthreads(); }
        float nrm = fmaxf(sqrtf(red[0]), 1e-12f);
        __syncthreads();
        if (t < 32) orow[128 + t] = v / nrm;
    }
    // segment 3: ego3 / ||ego3||  (16)
    {
        float v = (t < 16) ? ego3[(size_t)node * 16 + t] : 0.f;
        red[t] = v * v; __syncthreads();
        for (int s = 32; s > 0; s >>= 1) { if (t < s) red[t] += red[t + s]; __syncthreads(); }
        float nrm = fmaxf(sqrtf(red[0]), 1e-12f);
        if (t < 16) orow[160 + t] = v / nrm;
    }
}

// ---------------------------------------------------------------- launcher
extern "C" void kernel_launch(void* const* d_in, const int* in_sizes, int n_in,
                              void* d_out, int out_size, void* d_ws, size_t ws_size,
                              hipStream_t stream)
{
    const int*   users = (const int*)d_in[0];
    const int*   pos   = (const int*)d_in[1];
    const int*   neg   = (const int*)d_in[2];
    const int*   all_h = (const int*)d_in[3];
    const int*   all_t = (const int*)d_in[4];
    const float* av    = (const float*)d_in[5];
    const float* uemb  = (const float*)d_in[6];
    const float* eemb  = (const float*)d_in[7];
    const float* Wgc0  = (const float*)d_in[8];  const float* bgc0 = (const float*)d_in[9];
    const float* Wbi0  = (const float*)d_in[10]; const float* bbi0 = (const float*)d_in[11];
    const float* Wgc1  = (const float*)d_in[12]; const float* bgc1 = (const float*)d_in[13];
    const float* Wbi1  = (const float*)d_in[14]; const float* bbi1 = (const float*)d_in[15];
    const float* Wgc2  = (const float*)d_in[16]; const float* bgc2 = (const float*)d_in[17];
    const float* Wbi2  = (const float*)d_in[18]; const float* bbi2 = (const float*)d_in[19];

    float* ws   = (float*)d_ws;
    float* side = ws;                                 // 150000*64
    float* ego1 = side + (size_t)N_NODES * 64;        // 150000*64
    float* ego2 = ego1 + (size_t)N_NODES * 64;        // 150000*32
    float* ego3 = ego2 + (size_t)N_NODES * 32;        // 150000*16
    float* out  = (float*)d_out;

    auto zero = [&](float* p, long n) {
        long n4 = n / 4;
        int blocks = (int)((n4 + 255) / 256);
        zero4_kernel<<<blocks, 256, 0, stream>>>((float4*)p, n4);
    };

    const int GEMM_BLOCKS = (N_NODES / 16 + 7) / 8;   // 8 waves/block

    // ---- layer 0: din 64 (concat embed), dout 64
    zero(side, (long)N_NODES * 64);
    {
        long total = (long)N_EDGES * 16;
        int blocks = (int)((total + 255) / 256);
        edge_scatter<64, true><<<blocks, 256, 0, stream>>>(
            all_h, all_t, av, nullptr, uemb, eemb, side);
    }
    gemm_bi<64, 64, true><<<GEMM_BLOCKS, 256, 0, stream>>>(
        nullptr, uemb, eemb, side, Wgc0, bgc0, Wbi0, bbi0, ego1);

    // ---- layer 1: din 64, dout 32
    zero(side, (long)N_NODES * 64);
    {
        long total = (long)N_EDGES * 16;
        int blocks = (int)((total + 255) / 256);
        edge_scatter<64, false><<<blocks, 256, 0, stream>>>(
            all_h, all_t, av, ego1, nullptr, nullptr, side);
    }
    gemm_bi<64, 32, false><<<GEMM_BLOCKS, 256, 0, stream>>>(
        ego1, nullptr, nullptr, side, Wgc1, bgc1, Wbi1, bbi1, ego2);

    // ---- layer 2: din 32, dout 16
    zero(side, (long)N_NODES * 32);
    {
        long total = (long)N_EDGES * 8;
        int blocks = (int)((total + 255) / 256);
        edge_scatter<32, false><<<blocks, 256, 0, stream>>>(
            all_h, all_t, av, ego2, nullptr, nullptr, side);
    }
    gemm_bi<32, 16, false><<<GEMM_BLOCKS, 256, 0, stream>>>(
        ego2, nullptr, nullptr, side, Wgc2, bgc2, Wbi2, bbi2, ego3);

    // ---- batch gather with fused per-row L2 normalization
    gather_out_kernel<<<3 * BATCH, 64, 0, stream>>>(
        users, pos, neg, uemb, eemb, ego1, ego2, ego3, out);
}